// SlidingWindowEnergyNormalizer_53953379173290
// MI455X (gfx1250) — compile-verified
//
#include <hip/hip_runtime.h>
#include <hip/hip_bf16.h>

// Problem constants (from the reference)
#define BB   16
#define FF   257
#define TT   8000
#define WIN  20
#define EPSV 1e-8f

#define TILE   256           // output t's per block in K1
#define GROUPS 18            // 18*16 = 288 frame-sum columns (256 + 19 lookback, rounded up)
#define TBLK   ((TT + TILE - 1) / TILE)   // 32 t-tiles per batch

typedef __attribute__((ext_vector_type(2))) float v2f;
typedef __attribute__((ext_vector_type(4))) float v4f;
typedef __attribute__((ext_vector_type(8))) float v8f;

#if defined(__gfx1250__) && __has_builtin(__builtin_amdgcn_wmma_f32_16x16x4_f32)
#define HAVE_WMMA4 1
#else
#define HAVE_WMMA4 0
#endif

// -----------------------------------------------------------------------------
// K1: frame_sum over F via WMMA (ones-vector matmul), then sliding-window mean.
// One block handles 256 consecutive t for one batch b. 256 threads = 8 waves.
// Each wave owns groups of 16 t-columns; per group it accumulates
// D = ones(16x4) @ mag_chunk(4x16) + D over 64 chunks of 4 F-rows (K=4 WMMA),
// so every lane's D[0] holds the column sum for t-column (lane % 16).
// -----------------------------------------------------------------------------
__global__ void __launch_bounds__(256)
k1_framesum_mean(const float* __restrict__ mag, const float* __restrict__ bias,
                 float* __restrict__ meanOut, float* __restrict__ invOut) {
    __shared__ float fs[GROUPS * 16];   // frame_sum for t in [t0-19, t0+269)

    const int blk  = blockIdx.x;
    const int b    = blk / TBLK;
    const int t0   = (blk % TBLK) * TILE;
    const int tid  = threadIdx.x;
    const int wave = tid >> 5;
    const int lane = tid & 31;
    const int n    = lane & 15;          // t-column within group
    const int kh   = (lane >> 4) << 1;   // k half: lanes 0-15 -> k{0,1}, lanes 16-31 -> k{2,3}

    const float* __restrict__ magb = mag + (size_t)b * FF * TT;

    // Uniform per-wave loop: EXEC stays all-ones through the WMMA region.
    for (int g = wave; g < GROUPS; g += 8) {
        const int   t     = t0 - (WIN - 1) + g * 16 + n;
        const bool  valid = (t >= 0) && (t < TT);
        const int   tc    = valid ? t : 0;          // safe address
        const float msk   = valid ? 1.0f : 0.0f;    // branchless zero for OOB cols
        const float* __restrict__ col = magb + tc;

        float sum;
#if HAVE_WMMA4
        v2f a; a.x = 1.0f; a.y = 1.0f;              // A = ones(16x4)
        v8f acc = {};
#pragma unroll 4
        for (int f0 = 0; f0 < 256; f0 += 4) {
            v2f bm;
            bm.x = col[(size_t)(f0 + kh)     * TT] * msk;
            bm.y = col[(size_t)(f0 + kh + 1) * TT] * msk;
            // (neg_a, A, neg_b, B, c_mod, C, reuse_a, reuse_b)
            acc = __builtin_amdgcn_wmma_f32_16x16x4_f32(
                false, a, false, bm, (short)0, acc, false, false);
        }
        sum = acc[0];                                // all D rows identical; n = lane%16
#else
        sum = 0.0f;
        for (int f = 0; f < 256; ++f) sum += col[(size_t)f * TT] * msk;
#endif
        sum += col[(size_t)256 * TT] * msk;          // leftover F row (257 = 64*4 + 1)
        if (lane < 16) fs[g * 16 + n] = sum;
    }
    __syncthreads();

    // Sliding-window mean: window for t=t0+j is fs[j .. j+19] (OOB entries are 0).
    const int t = t0 + tid;
    if (t < TT) {
        float w = 0.0f;
#pragma unroll
        for (int i = 0; i < WIN; ++i) w += fs[tid + i];
        const int   cnt = (t + 1 < WIN ? t + 1 : WIN) * FF;
        const float m   = w / (float)cnt + bias[0];
        meanOut[b * TT + t] = m;
        if (invOut) invOut[b * TT + t] = 1.0f / (m + EPSV);   // 128K divides total: negligible
    }
}

// -----------------------------------------------------------------------------
// K2: norm = mag * inv(mean+eps). Streaming, b128 (ext_vector float4) loads and
// non-temporal b128 stores. mag should hit the 192MB L2 (populated by K1); the
// NT norm stores avoid evicting not-yet-read mag lines.
// grid: x covers t/4 (8 blocks * 256 thr = 2048 >= 2000), y = row = b*FF + f.
// -----------------------------------------------------------------------------
__global__ void __launch_bounds__(256)
k2_normalize(const float* __restrict__ mag, const float* __restrict__ meanIn,
             const float* __restrict__ invIn, float* __restrict__ out) {
    const int t4  = (blockIdx.x * blockDim.x + threadIdx.x) * 4;
    if (t4 >= TT) return;
    const int row = blockIdx.y;          // b*FF + f
    const int b   = row / FF;

    const size_t e = (size_t)row * TT + t4;
    const v4f m = *reinterpret_cast<const v4f*>(mag + e);

    v4f iv;
    if (invIn) {
        iv = *reinterpret_cast<const v4f*>(invIn + (size_t)b * TT + t4);
    } else {
        const v4f mn = *reinterpret_cast<const v4f*>(meanIn + (size_t)b * TT + t4);
        iv.x = 1.0f / (mn.x + EPSV);
        iv.y = 1.0f / (mn.y + EPSV);
        iv.z = 1.0f / (mn.z + EPSV);
        iv.w = 1.0f / (mn.w + EPSV);
    }
    const v4f r = m * iv;
    __builtin_nontemporal_store(r, reinterpret_cast<v4f*>(out + e));
}

extern "C" void kernel_launch(void* const* d_in, const int* in_sizes, int n_in,
                              void* d_out, int out_size, void* d_ws, size_t ws_size,
                              hipStream_t stream) {
    (void)in_sizes; (void)n_in; (void)out_size;

    const float* mag  = (const float*)d_in[0];
    const float* bias = (const float*)d_in[1];
    float*       out  = (float*)d_out;
    float*       mean = out + (size_t)BB * FF * TT;   // outputs: norm flat, then mean

    float* inv = nullptr;
    if (d_ws && ws_size >= (size_t)BB * TT * sizeof(float))
        inv = (float*)d_ws;

    k1_framesum_mean<<<BB * TBLK, 256, 0, stream>>>(mag, bias, mean, inv);

    dim3 g2((TT / 4 + 255) / 256, BB * FF);            // (8, 4112)
    k2_normalize<<<g2, 256, 0, stream>>>(mag, mean, inv, out);
}